// HybridFCLLSTM_65481071402083
// MI455X (gfx1250) — compile-verified
//
#include <hip/hip_runtime.h>

// ---------------------------------------------------------------------------
// Problem constants (from reference)
// ---------------------------------------------------------------------------
static constexpr int SEQ    = 1024;
static constexpr int BATCH  = 2048;
static constexpr int TAGSET = 16;
static constexpr int NW     = 4;   // n_wires / hidden dim
static constexpr int IN_DIM = 5;   // [x, h0..h3]

typedef __attribute__((ext_vector_type(2))) float v2f;
typedef __attribute__((ext_vector_type(8))) float v8f;

// ---------------------------------------------------------------------------
// Cross-lane helpers: immediate-mode ds_swizzle (no address VGPR, no LDS mem)
//   quad-perm broadcasts (offset[15]=1):  0x8000|sel3<<6|sel2<<4|sel1<<2|sel0
//   bitmask-perm xor     (offset[15]=0):  xor<<10 | or<<5 | and
// ---------------------------------------------------------------------------
template <int IMM>
__device__ __forceinline__ float swz(float x) {
    return __int_as_float(__builtin_amdgcn_ds_swizzle(__float_as_int(x), IMM));
}
static constexpr int QB0 = 0x8000;  // quad broadcast lane 0 of each 4-group
static constexpr int QB1 = 0x8055;  // quad broadcast lane 1
static constexpr int QB2 = 0x80AA;  // quad broadcast lane 2
static constexpr int QB3 = 0x80FF;  // quad broadcast lane 3
static constexpr int X1  = 0x041F;  // xor 1, and 0x1f
static constexpr int X2  = 0x081F;  // xor 2
static constexpr int X4  = 0x101F;  // xor 4
static constexpr int X8  = 0x201F;  // xor 8

// ---------------------------------------------------------------------------
// Fast device math (native v_sin/v_cos/v_exp/v_log/v_rcp)
// ---------------------------------------------------------------------------
__device__ __forceinline__ float fast_sigmoid(float x) {
    return 1.0f / (1.0f + __expf(-x));
}
__device__ __forceinline__ float fast_tanh(float x) {
    float e = __expf(2.0f * x);          // exact at +/-inf limits
    return 1.0f - 2.0f / (e + 1.0f);
}

// ---------------------------------------------------------------------------
// Kernel 1: serial scan, 4 lanes per batch chain (one lane per wire).
//   * per-lane transcendental load drops ~4x vs 1-lane-per-chain
//   * gate cosines + hidden state exchanged with quad-perm ds_swizzle
//     (16 swizzles -> 1 dscnt sync; 4 swizzles -> 1 dscnt sync per step)
//   * 256 single-wave workgroups -> 4x occupancy of the 1-lane version
//   * hs store: 1 dword/lane, 128B contiguous per wave
// CNOT-ring expectations per wire j (c = cos(z+phi)):
//   y0=c1*c2*c3, y1=c0*c1, y2=c0*c1*c2, y3=c0*c1*c2*c3
//   -> y_j = (j==0?1:c0) * c1 * (j==1?1:c2) * (j==1||j==2?1:c3)
// ---------------------------------------------------------------------------
__global__ __launch_bounds__(32) void qlstm_scan4(
    const int* __restrict__ sent, const float* __restrict__ emb,
    const float* __restrict__ Wf, const float* __restrict__ bf,
    const float* __restrict__ Wi, const float* __restrict__ bi,
    const float* __restrict__ Wg, const float* __restrict__ bg,
    const float* __restrict__ Wo, const float* __restrict__ bo,
    const float* __restrict__ phf, const float* __restrict__ phi,
    const float* __restrict__ phg, const float* __restrict__ pho,
    float* __restrict__ hs) {
    const int lane  = threadIdx.x & 31;
    const int sub   = lane & 3;                    // wire id
    const int chain = blockIdx.x * 8 + (lane >> 2); // batch element

    // This lane's weight columns (wire = sub); b and phi fold into one const:
    // cos((comb@W + b) + phi) = cos(comb@W + (b+phi))
    float wF[IN_DIM], wI[IN_DIM], wG[IN_DIM], wO[IN_DIM];
#pragma unroll
    for (int i = 0; i < IN_DIM; ++i) {
        wF[i] = Wf[i * NW + sub];
        wI[i] = Wi[i * NW + sub];
        wG[i] = Wg[i * NW + sub];
        wO[i] = Wo[i * NW + sub];
    }
    const float bF = bf[sub] + phf[sub];
    const float bI = bi[sub] + phi[sub];
    const float bG = bg[sub] + phg[sub];
    const float bO = bo[sub] + pho[sub];

    // Loop-invariant lane masks for the ring-product selects.
    const bool m0  = (sub == 0);
    const bool m1  = (sub == 1);
    const bool m12 = (sub == 1) || (sub == 2);

    float h0 = 0.f, h1 = 0.f, h2 = 0.f, h3 = 0.f;  // replicated hidden state
    float cs = 0.f;                                 // this wire's cell state

    for (int t = 0; t < SEQ; ++t) {
        const int tok = sent[(size_t)t * BATCH + chain];
        const float x = fmaf(__sinf(emb[tok]), 0.5f, 0.5f);

        // z per gate for this wire (5 FMA each), then cosine.
        float zf = bF, zi = bI, zg = bG, zo = bO;
        zf = fmaf(x, wF[0], zf); zi = fmaf(x, wI[0], zi);
        zg = fmaf(x, wG[0], zg); zo = fmaf(x, wO[0], zo);
        zf = fmaf(h0, wF[1], zf); zi = fmaf(h0, wI[1], zi);
        zg = fmaf(h0, wG[1], zg); zo = fmaf(h0, wO[1], zo);
        zf = fmaf(h1, wF[2], zf); zi = fmaf(h1, wI[2], zi);
        zg = fmaf(h1, wG[2], zg); zo = fmaf(h1, wO[2], zo);
        zf = fmaf(h2, wF[3], zf); zi = fmaf(h2, wI[3], zi);
        zg = fmaf(h2, wG[3], zg); zo = fmaf(h2, wO[3], zo);
        zf = fmaf(h3, wF[4], zf); zi = fmaf(h3, wI[4], zi);
        zg = fmaf(h3, wG[4], zg); zo = fmaf(h3, wO[4], zo);

        const float ccf = __cosf(zf);
        const float cci = __cosf(zi);
        const float ccg = __cosf(zg);
        const float cco = __cosf(zo);

        // Broadcast all wires' cosines per gate (16 swizzles, one dscnt sync).
        const float f0 = swz<QB0>(ccf), f1 = swz<QB1>(ccf), f2 = swz<QB2>(ccf), f3 = swz<QB3>(ccf);
        const float i0 = swz<QB0>(cci), i1 = swz<QB1>(cci), i2 = swz<QB2>(cci), i3 = swz<QB3>(cci);
        const float g0 = swz<QB0>(ccg), g1 = swz<QB1>(ccg), g2 = swz<QB2>(ccg), g3 = swz<QB3>(ccg);
        const float o0 = swz<QB0>(cco), o1 = swz<QB1>(cco), o2 = swz<QB2>(cco), o3 = swz<QB3>(cco);

        // Ring products via masked factors (3 selects + 3 muls per gate).
        const float yf = (m0 ? 1.0f : f0) * f1 * (m1 ? 1.0f : f2) * (m12 ? 1.0f : f3);
        const float yi = (m0 ? 1.0f : i0) * i1 * (m1 ? 1.0f : i2) * (m12 ? 1.0f : i3);
        const float yg = (m0 ? 1.0f : g0) * g1 * (m1 ? 1.0f : g2) * (m12 ? 1.0f : g3);
        const float yo = (m0 ? 1.0f : o0) * o1 * (m1 ? 1.0f : o2) * (m12 ? 1.0f : o3);

        const float fj = fast_sigmoid(yf);
        const float ij = fast_sigmoid(yi);
        const float gj = fast_tanh(yg);
        const float oj = fast_sigmoid(yo);

        cs = fmaf(fj, cs, ij * gj);
        const float hj = oj * fast_tanh(cs);

        // Coalesced: lanes of the wave cover 32 consecutive floats.
        hs[((size_t)t * BATCH + chain) * NW + sub] = hj;

        // Re-replicate hidden state for the next step (4 swizzles, one sync).
        h0 = swz<QB0>(hj); h1 = swz<QB1>(hj); h2 = swz<QB2>(hj); h3 = swz<QB3>(hj);
    }
}

// ---------------------------------------------------------------------------
// Kernel 2: classifier + log_softmax via V_WMMA_F32_16X16X4_F32.
//   A (16x4 f32)  = 16 rows of hs   (lane L: row L&15, K = 2*(L>>4)+{0,1})
//   B (4x16 f32)  = Wt              (lane L: col L&15, K = 2*(L>>4)+{0,1})
//   C (16x16 f32) = bias bt broadcast per column
//   D row M=r -> lanes 0-15 of d[r]; row M=8+r -> lanes 16-31 of d[r]
// Row-wise log-softmax = 16-lane xor butterflies via immediate ds_swizzle.
// ---------------------------------------------------------------------------
__global__ __launch_bounds__(256) void qlstm_logits_wmma(
    const float* __restrict__ hs, const float* __restrict__ Wt,
    const float* __restrict__ bt, float* __restrict__ out, int nrows) {
    const int gtid    = blockIdx.x * blockDim.x + threadIdx.x;
    const int wave    = gtid >> 5;          // one 16-row tile per wave
    const int lane    = threadIdx.x & 31;
    const int rowBase = wave * 16;
    if (rowBase >= nrows) return;           // wave-uniform: EXEC stays all-1s

    const int n  = lane & 15;               // column / N index
    const int kh = lane >> 4;               // K half (0 -> K{0,1}, 1 -> K{2,3})

    // A: 8 contiguous bytes per lane; 256B contiguous per wave.
    const v2f* hs2 = reinterpret_cast<const v2f*>(hs);
    v2f a = hs2[(size_t)(rowBase + n) * 2 + kh];

    // B: Wt is (4 x 16) row-major.
    v2f bm;
    bm.x = Wt[(2 * kh + 0) * TAGSET + n];
    bm.y = Wt[(2 * kh + 1) * TAGSET + n];

    const float bias = bt[n];
    v8f cacc;
#pragma unroll
    for (int r = 0; r < 8; ++r) cacc[r] = bias;

    v8f d = __builtin_amdgcn_wmma_f32_16x16x4_f32(
        /*neg_a=*/false, a, /*neg_b=*/false, bm,
        /*c_mod=*/(short)0, cacc, /*reuse_a=*/false, /*reuse_b=*/false);

    // log_softmax over each 16-wide row (one row per half-wave per register).
#pragma unroll
    for (int r = 0; r < 8; ++r) {
        float v = d[r];
        float m = v;
        m = fmaxf(m, swz<X1>(m));
        m = fmaxf(m, swz<X2>(m));
        m = fmaxf(m, swz<X4>(m));
        m = fmaxf(m, swz<X8>(m));
        float s = __expf(v - m);
        s += swz<X1>(s);
        s += swz<X2>(s);
        s += swz<X4>(s);
        s += swz<X8>(s);
        const float res = v - m - __logf(s);
        const int rr = rowBase + r + (kh << 3);     // global output row
        out[(size_t)rr * TAGSET + n] = res;         // 2x64B coalesced segments
    }
}

// ---------------------------------------------------------------------------
// Fallback (only if ws_size < 32MB): fused scan emitting log-softmax directly.
// ---------------------------------------------------------------------------
__device__ __forceinline__ void qlayer4_all(const float comb[IN_DIM],
                                            const float W[IN_DIM * NW],
                                            const float b[NW],
                                            const float ph[NW],
                                            float y[NW]) {
    float cz[NW];
#pragma unroll
    for (int j = 0; j < NW; ++j) {
        float z = b[j];
#pragma unroll
        for (int i = 0; i < IN_DIM; ++i) z = fmaf(comb[i], W[i * NW + j], z);
        cz[j] = __cosf(z + ph[j]);
    }
    float p01  = cz[0] * cz[1];
    float p012 = p01 * cz[2];
    y[0] = cz[1] * cz[2] * cz[3];
    y[1] = p01;
    y[2] = p012;
    y[3] = p012 * cz[3];
}

__global__ __launch_bounds__(32) void qlstm_scan_fused(
    const int* __restrict__ sent, const float* __restrict__ emb,
    const float* __restrict__ Wf, const float* __restrict__ bf,
    const float* __restrict__ Wi, const float* __restrict__ bi,
    const float* __restrict__ Wg, const float* __restrict__ bg,
    const float* __restrict__ Wo, const float* __restrict__ bo,
    const float* __restrict__ phf, const float* __restrict__ phi,
    const float* __restrict__ phg, const float* __restrict__ pho,
    const float* __restrict__ Wt, const float* __restrict__ bt,
    float* __restrict__ out) {
    const int b = blockIdx.x * 32 + (threadIdx.x & 31);
    if (b >= BATCH) return;

    float Wf_[20], Wi_[20], Wg_[20], Wo_[20];
#pragma unroll
    for (int k = 0; k < 20; ++k) {
        Wf_[k] = Wf[k]; Wi_[k] = Wi[k]; Wg_[k] = Wg[k]; Wo_[k] = Wo[k];
    }
    float bf_[4], bi_[4], bg_[4], bo_[4], pf_[4], pi_[4], pg_[4], po_[4];
#pragma unroll
    for (int k = 0; k < 4; ++k) {
        bf_[k] = bf[k]; bi_[k] = bi[k]; bg_[k] = bg[k]; bo_[k] = bo[k];
        pf_[k] = phf[k]; pi_[k] = phi[k]; pg_[k] = phg[k]; po_[k] = pho[k];
    }
    float Wt_[NW * TAGSET], bt_[TAGSET];
#pragma unroll
    for (int k = 0; k < NW * TAGSET; ++k) Wt_[k] = Wt[k];
#pragma unroll
    for (int k = 0; k < TAGSET; ++k) bt_[k] = bt[k];

    float h[NW] = {0.f, 0.f, 0.f, 0.f};
    float c[NW] = {0.f, 0.f, 0.f, 0.f};

    for (int t = 0; t < SEQ; ++t) {
        const int tok = sent[(size_t)t * BATCH + b];
        const float x = fmaf(__sinf(emb[tok]), 0.5f, 0.5f);
        const float comb[IN_DIM] = {x, h[0], h[1], h[2], h[3]};
        float fg[NW], ig[NW], gg[NW], og[NW];
        qlayer4_all(comb, Wf_, bf_, pf_, fg);
        qlayer4_all(comb, Wi_, bi_, pi_, ig);
        qlayer4_all(comb, Wg_, bg_, pg_, gg);
        qlayer4_all(comb, Wo_, bo_, po_, og);
#pragma unroll
        for (int j = 0; j < NW; ++j) {
            const float f = fast_sigmoid(fg[j]);
            const float i = fast_sigmoid(ig[j]);
            const float g = fast_tanh(gg[j]);
            const float o = fast_sigmoid(og[j]);
            const float cc = fmaf(f, c[j], i * g);
            c[j] = cc;
            h[j] = o * fast_tanh(cc);
        }
        float lg[TAGSET];
        float m = -3.402823e38f;
#pragma unroll
        for (int j = 0; j < TAGSET; ++j) {
            float z = bt_[j];
#pragma unroll
            for (int k = 0; k < NW; ++k) z = fmaf(h[k], Wt_[k * TAGSET + j], z);
            lg[j] = z;
            m = fmaxf(m, z);
        }
        float s = 0.f;
#pragma unroll
        for (int j = 0; j < TAGSET; ++j) s += __expf(lg[j] - m);
        const float ls = __logf(s);
        const size_t base = ((size_t)t * BATCH + b) * TAGSET;
#pragma unroll
        for (int j = 0; j < TAGSET; ++j) out[base + j] = lg[j] - m - ls;
    }
}

// ---------------------------------------------------------------------------
// Launch
// ---------------------------------------------------------------------------
extern "C" void kernel_launch(void* const* d_in, const int* in_sizes, int n_in,
                              void* d_out, int out_size, void* d_ws, size_t ws_size,
                              hipStream_t stream) {
    const int*   sent = (const int*)  d_in[0];
    const float* emb  = (const float*)d_in[1];
    const float* Wf   = (const float*)d_in[2];
    const float* bf   = (const float*)d_in[3];
    const float* Wi   = (const float*)d_in[4];
    const float* bi   = (const float*)d_in[5];
    const float* Wg   = (const float*)d_in[6];
    const float* bg   = (const float*)d_in[7];
    const float* Wo   = (const float*)d_in[8];
    const float* bo   = (const float*)d_in[9];
    const float* phf  = (const float*)d_in[10];
    const float* phi  = (const float*)d_in[11];
    const float* phg  = (const float*)d_in[12];
    const float* pho  = (const float*)d_in[13];
    const float* Wt   = (const float*)d_in[14];
    const float* bt   = (const float*)d_in[15];
    float* out = (float*)d_out;

    const size_t hs_bytes = (size_t)SEQ * BATCH * NW * sizeof(float);
    if (ws_size >= hs_bytes) {
        float* hs = (float*)d_ws;
        // 4 lanes per chain, 8 chains per wave, 256 single-wave workgroups.
        qlstm_scan4<<<BATCH / 8, 32, 0, stream>>>(
            sent, emb, Wf, bf, Wi, bi, Wg, bg, Wo, bo, phf, phi, phg, pho, hs);

        const int rows  = SEQ * BATCH;          // 2,097,152
        const int tiles = rows / 16;            // 131,072 waves, 1 WMMA each
        qlstm_logits_wmma<<<tiles / 8, 256, 0, stream>>>(hs, Wt, bt, out, rows);
    } else {
        qlstm_scan_fused<<<BATCH / 32, 32, 0, stream>>>(
            sent, emb, Wf, bf, Wi, bi, Wg, bg, Wo, bo, phf, phi, phg, pho,
            Wt, bt, out);
    }
}